// YOLOv3Loss_18193481466543
// MI455X (gfx1250) — compile-verified
//
#include <hip/hip_runtime.h>
#include <math.h>
#include <stdint.h>

#define BATCH 16
#define KGT   50
#define NA    3
#define NC    80
#define IGNORE_T 0.7f

typedef float v2f __attribute__((ext_vector_type(2)));
typedef float v8f __attribute__((ext_vector_type(8)));

#if defined(__gfx1250__) && __has_builtin(__builtin_amdgcn_wmma_f32_16x16x4_f32)
#define USE_WMMA_RED 1
#endif

#if defined(__gfx1250__) && __has_builtin(__builtin_amdgcn_global_load_async_to_lds_b32) && __has_builtin(__builtin_amdgcn_s_wait_asynccnt)
#define USE_ASYNC_LDS 1
#endif

__device__ __forceinline__ float sigf(float x) { return 1.0f / (1.0f + expf(-x)); }

// Block-wide sum. Wave level uses V_WMMA_F32_16X16X4_F32 with A == all-ones:
// D[m][n] = sum_k B[k][n]; with B = (v, 0) each column sum is v[n] + v[n+16],
// so d[0] lanes 0..15 hold 16 pair-sums of the wave (exact: x*1.0 is exact).
// Result valid in thread 0 only. Must be called by ALL threads of the block
// with no divergent return before it (WMMA requires EXEC all-ones).
template <int BS>
__device__ __forceinline__ float block_sum(float v) {
    __shared__ float sdata[(BS / 32) * 16];
    const int lane = threadIdx.x & 31;
    const int wave = threadIdx.x >> 5;
#ifdef USE_WMMA_RED
    v2f a; a.x = 1.0f; a.y = 1.0f;
    v2f bm; bm.x = v; bm.y = 0.0f;
    v8f c = {};
    v8f d = __builtin_amdgcn_wmma_f32_16x16x4_f32(false, a, false, bm,
                                                  (short)0, c, false, false);
    float s = d[0];
#else
    float s = v + __shfl_xor(v, 16, 32);
#endif
    if (lane < 16) sdata[wave * 16 + lane] = s;
    __syncthreads();
    float tot = 0.0f;
    if (threadIdx.x == 0) {
        for (int i = 0; i < (BS / 32) * 16; ++i) tot += sdata[i];
    }
    __syncthreads();
    return tot;
}

// ---------------------------------------------------------------------------
// Kernel 1: per-GT prep. One block per batch. Computes responsible cell,
// anchor argmax (first-max tie rule), delta, winner flag (= last valid GT
// mapping to the same (cell, anchor) slot; replaces the segment_max scatter),
// num_obj, and clears has_pos.
// ---------------------------------------------------------------------------
__global__ void yolo_prep(const float* __restrict__ targets,
                          const float* __restrict__ anchors,
                          int W, int H,
                          int* __restrict__ has_pos, int* __restrict__ num_obj,
                          int* __restrict__ wpos, int* __restrict__ wcls,
                          float* __restrict__ wdelta) {
    const int b = blockIdx.x;
    const int k = threadIdx.x;
    __shared__ int s_f[KGT];

    bool valid = false;
    int f_local = -1;
    float d0 = 0.f, d1 = 0.f, d2 = 0.f, d3 = 0.f;
    int cls = 0;

    if (k < KGT) {
        const float* t = targets + ((size_t)b * KGT + k) * 5;
        float tx = t[0], ty = t[1], tw = t[2], th = t[3], tc = t[4];
        valid = (tx + ty + tw + th + tc) > 0.0f;
        float gx = tx * (float)W, gy = ty * (float)H;
        float gw = tw * (float)W, gh = th * (float)H;
        cls = (int)tc;
        int cx = (int)floorf(gx); cx = cx < 0 ? 0 : (cx > W - 1 ? W - 1 : cx);
        int cy = (int)floorf(gy); cy = cy < 0 ? 0 : (cy > H - 1 ? H - 1 : cy);
        float gx1 = gx - gw * 0.5f, gx2 = gx + gw * 0.5f;
        float gy1 = gy - gh * 0.5f, gy2 = gy + gh * 0.5f;
        float garea = gw * gh;
        float acx = (float)cx + 0.5f, acy = (float)cy + 0.5f;
        float best = -1e30f;
        int a_star = 0;
        for (int a = 0; a < NA; ++a) {
            float aw = anchors[2 * a], ah = anchors[2 * a + 1];
            float iw = fminf(acx + aw * 0.5f, gx2) - fmaxf(acx - aw * 0.5f, gx1);
            float ih = fminf(acy + ah * 0.5f, gy2) - fmaxf(acy - ah * 0.5f, gy1);
            iw = fmaxf(iw, 0.0f); ih = fmaxf(ih, 0.0f);
            float inter = iw * ih;
            float ov = inter / (aw * ah + garea - inter);
            if (ov > best) { best = ov; a_star = a; }
        }
        float aw = anchors[2 * a_star], ah = anchors[2 * a_star + 1];
        d0 = gx - (float)cx; d1 = gy - (float)cy;
        d2 = gw / aw;        d3 = gh / ah;
        f_local = (cy * W + cx) * NA + a_star;
        s_f[k] = valid ? f_local : -1;
    }

    int cnt = __syncthreads_count((int)((k < KGT) && valid));
    if (threadIdx.x == 0) { num_obj[b] = cnt; has_pos[b] = 0; }

    if (k < KGT) {
        bool win = valid;
        if (win) {
            for (int k2 = k + 1; k2 < KGT; ++k2)
                if (s_f[k2] == f_local) { win = false; break; }
        }
        int i = b * KGT + k;
        wpos[i] = win ? (b * H * W * NA + f_local) : -1;
        wcls[i] = cls;
        float* wd = wdelta + 4 * i;
        wd[0] = d0; wd[1] = d1; wd[2] = d2; wd[3] = d3;
    }
}

// ---------------------------------------------------------------------------
// Kernel 2: max IoU per prediction vs 50 GTs (staged in LDS via gfx1250
// async-to-LDS when available), plus has_pos detection.
// Thread mapping: idx -> (a, hw) with hw fastest for coalesced channel reads.
// maxiou stored canonically at [b*HW*A + hw*A + a].
// ---------------------------------------------------------------------------
__global__ void yolo_maxiou(const float* __restrict__ o,
                            const float* __restrict__ targets,
                            const float* __restrict__ anchors,
                            int W, int H,
                            float* __restrict__ maxiou,
                            int* __restrict__ has_pos) {
    const int b = blockIdx.y;
    const int HW = H * W;
    const int tid = threadIdx.x;

    __shared__ float s_raw[KGT * 5];
    __shared__ float s_x1[KGT], s_y1[KGT], s_x2[KGT], s_y2[KGT], s_ar[KGT];
    __shared__ int   s_v[KGT];

    const float* tb = targets + (size_t)b * KGT * 5;
#ifdef USE_ASYNC_LDS
    if (tid < KGT * 5) {
        __builtin_amdgcn_global_load_async_to_lds_b32(
            (__attribute__((address_space(1))) int*)(uintptr_t)(tb + tid),
            (__attribute__((address_space(3))) int*)(uint32_t)(uintptr_t)(&s_raw[tid]),
            0, 0);
    }
    __builtin_amdgcn_s_wait_asynccnt(0);
    __syncthreads();
#else
    if (tid < KGT * 5) s_raw[tid] = tb[tid];
    __syncthreads();
#endif

    if (tid < KGT) {
        float tx = s_raw[tid * 5 + 0], ty = s_raw[tid * 5 + 1];
        float tw = s_raw[tid * 5 + 2], th = s_raw[tid * 5 + 3];
        float tc = s_raw[tid * 5 + 4];
        s_v[tid] = (tx + ty + tw + th + tc) > 0.0f;
        float gx = tx * (float)W, gy = ty * (float)H;
        float gw = tw * (float)W, gh = th * (float)H;
        s_x1[tid] = gx - gw * 0.5f; s_x2[tid] = gx + gw * 0.5f;
        s_y1[tid] = gy - gh * 0.5f; s_y2[tid] = gy + gh * 0.5f;
        s_ar[tid] = gw * gh;
    }
    __syncthreads();

    int idx = blockIdx.x * blockDim.x + tid;
    if (idx >= NA * HW) return;
    int a = idx / HW, hw = idx % HW;
    int y = hw / W, x = hw % W;

    const float* base = o + ((size_t)b * 255 + (size_t)a * 85) * HW + hw;
    float o0 = base[0];
    float o1 = base[(size_t)HW];
    float o2 = base[2 * (size_t)HW];
    float o3 = base[3 * (size_t)HW];
    float aw = anchors[2 * a], ah = anchors[2 * a + 1];
    float px = sigf(o0) + (float)x;
    float py = sigf(o1) + (float)y;
    float pw = expf(o2) * aw;
    float ph = expf(o3) * ah;
    float px1 = px - pw * 0.5f, px2 = px + pw * 0.5f;
    float py1 = py - ph * 0.5f, py2 = py + ph * 0.5f;
    float parea = pw * ph;

    float maxi = -1.0f;
    for (int kk = 0; kk < KGT; ++kk) {
        if (!s_v[kk]) continue;
        float iw = fminf(px2, s_x2[kk]) - fmaxf(px1, s_x1[kk]);
        float ih = fminf(py2, s_y2[kk]) - fmaxf(py1, s_y1[kk]);
        iw = fmaxf(iw, 0.0f); ih = fmaxf(ih, 0.0f);
        float inter = iw * ih;
        maxi = fmaxf(maxi, inter / (parea + s_ar[kk] - inter));
    }
    maxiou[(size_t)b * NA * HW + (size_t)hw * NA + a] = maxi;
    if (maxi > IGNORE_T) atomicOr(&has_pos[b], 1);
}

// ---------------------------------------------------------------------------
// Kernel 3: default (no-object) iou-loss term for ALL positions:
// 0.5 * (sigmoid(conf) * mask)^2 ; mask = 0 if num_obj==0 or (has_pos &&
// max_iou >= 0.7), else 1. Winner positions are corrected in kernel 4.
// ---------------------------------------------------------------------------
__global__ void yolo_noobj(const float* __restrict__ o,
                           const float* __restrict__ maxiou,
                           const int* __restrict__ has_pos,
                           const int* __restrict__ num_obj,
                           int W, int H,
                           float* __restrict__ partial) {
    const int b = blockIdx.y;
    const int HW = H * W;
    int idx = blockIdx.x * blockDim.x + threadIdx.x;
    float val = 0.0f;
    if (idx < NA * HW && num_obj[b] > 0) {
        int a = idx / HW, hw = idx % HW;
        float mi = maxiou[(size_t)b * NA * HW + (size_t)hw * NA + a];
        bool ign = has_pos[b] && (mi >= IGNORE_T);
        if (!ign) {
            float conf = o[((size_t)b * 255 + (size_t)a * 85 + 4) * HW + hw];
            float c = sigf(conf);
            val = 0.5f * c * c;
        }
    }
    float tot = block_sum<256>(val);
    if (threadIdx.x == 0) partial[blockIdx.y * gridDim.x + blockIdx.x] = tot;
}

// ---------------------------------------------------------------------------
// Kernel 4: winner terms (<= B*K positions). Replaces the default iou term
// with the object term, adds box MSE and online-logsumexp class loss.
// ---------------------------------------------------------------------------
__global__ void yolo_winners(const float* __restrict__ o,
                             const float* __restrict__ maxiou,
                             const int* __restrict__ has_pos,
                             const int* __restrict__ num_obj,
                             const int* __restrict__ wpos,
                             const int* __restrict__ wcls,
                             const float* __restrict__ wdelta,
                             int W, int H,
                             float* __restrict__ partial) {
    const int i = blockIdx.x * blockDim.x + threadIdx.x;
    const int HW = H * W;
    float val = 0.0f;
    if (i < BATCH * KGT) {
        int pos = wpos[i];
        if (pos >= 0) {
            int b = pos / (NA * HW);
            int r = pos % (NA * HW);
            int cell = r / NA, a = r % NA;
            const float* base = o + ((size_t)b * 255 + (size_t)a * 85) * HW + cell;
            float o0 = base[0];
            float o1 = base[(size_t)HW];
            float o2 = base[2 * (size_t)HW];
            float o3 = base[3 * (size_t)HW];
            float o4 = base[4 * (size_t)HW];

            float conf = sigf(o4);
            float obj_iou = maxiou[pos];
            // remove the default contribution computed by yolo_noobj here
            float dm = (num_obj[b] > 0)
                           ? ((has_pos[b] && obj_iou >= IGNORE_T) ? 0.0f : 1.0f)
                           : 0.0f;
            float ds = conf * dm;
            val -= 0.5f * ds * ds;
            // object term: mask = OBJ_SCALE = 5
            float w = 5.0f * (conf - obj_iou);
            val += 0.5f * w * w;
            // box loss (COORD_SCALE/2 = 0.5)
            const float* dl = wdelta + 4 * i;
            float e0 = sigf(o0) - dl[0];
            float e1 = sigf(o1) - dl[1];
            float e2 = expf(o2) - dl[2];
            float e3 = expf(o3) - dl[3];
            val += 0.5f * (e0 * e0 + e1 * e1 + e2 * e2 + e3 * e3);
            // class loss: lse - target logit (online logsumexp, one pass)
            float m = -3.4e38f, ssum = 0.0f;
            for (int c = 0; c < NC; ++c) {
                float l = base[(size_t)(5 + c) * HW];
                if (l > m) { ssum = ssum * expf(m - l) + 1.0f; m = l; }
                else       { ssum += expf(l - m); }
            }
            float lse = m + logf(ssum);
            float tgt = base[(size_t)(5 + wcls[i]) * HW];
            val += lse - tgt;
        }
    }
    float tot = block_sum<128>(val);
    if (threadIdx.x == 0) partial[blockIdx.x] = tot;
}

// Deterministic fixed-order reduction of partials -> per-scale accumulator.
__global__ void yolo_reduce(const float* __restrict__ p1, int n1,
                            const float* __restrict__ p2, int n2,
                            float* __restrict__ acc, int s) {
    if (threadIdx.x == 0 && blockIdx.x == 0) {
        float sum = 0.0f;
        for (int i = 0; i < n1; ++i) sum += p1[i];
        for (int i = 0; i < n2; ++i) sum += p2[i];
        acc[s] = sum;
    }
}

__global__ void yolo_final(const float* __restrict__ acc, float* __restrict__ out) {
    if (threadIdx.x == 0 && blockIdx.x == 0)
        out[0] = (acc[0] + acc[1] + acc[2]) * (1.0f / ((float)BATCH * 3.0f));
}

extern "C" void kernel_launch(void* const* d_in, const int* in_sizes, int n_in,
                              void* d_out, int out_size, void* d_ws, size_t ws_size,
                              hipStream_t stream) {
    const float* o_arr[3] = {(const float*)d_in[0], (const float*)d_in[1],
                             (const float*)d_in[2]};
    const float* targets = (const float*)d_in[3];
    const float* anchors = (const float*)d_in[4];

    const int Hs[3]   = {13, 26, 52};  // o1, o2, o3
    const int aoff[3] = {6, 3, 0};     // anchors[6:9], [3:6], [0:3]

    char* ws = (char*)d_ws;
    const int NMAX = BATCH * 52 * 52 * NA;  // largest scale
    float* maxiou  = (float*)ws; ws += (size_t)NMAX * 4;
    int*   has_pos = (int*)ws;   ws += BATCH * 4;
    int*   num_obj = (int*)ws;   ws += BATCH * 4;
    int*   wpos    = (int*)ws;   ws += BATCH * KGT * 4;
    int*   wcls    = (int*)ws;   ws += BATCH * KGT * 4;
    float* wdelta  = (float*)ws; ws += BATCH * KGT * 4 * 4;
    float* partial = (float*)ws; ws += 1024 * 4;
    float* pwin    = (float*)ws; ws += 16 * 4;
    float* acc     = (float*)ws; ws += 4 * 4;

    for (int s = 0; s < 3; ++s) {
        const int H = Hs[s], W = H, HW = H * W;
        const float* anc = anchors + aoff[s] * 2;
        const float* o   = o_arr[s];

        yolo_prep<<<BATCH, 64, 0, stream>>>(targets, anc, W, H, has_pos,
                                            num_obj, wpos, wcls, wdelta);

        const int gx = (NA * HW + 255) / 256;
        dim3 g2(gx, BATCH);
        yolo_maxiou<<<g2, 256, 0, stream>>>(o, targets, anc, W, H, maxiou,
                                            has_pos);
        yolo_noobj<<<g2, 256, 0, stream>>>(o, maxiou, has_pos, num_obj, W, H,
                                           partial);

        const int gw = (BATCH * KGT + 127) / 128;
        yolo_winners<<<gw, 128, 0, stream>>>(o, maxiou, has_pos, num_obj, wpos,
                                             wcls, wdelta, W, H, pwin);

        yolo_reduce<<<1, 1, 0, stream>>>(partial, BATCH * gx, pwin, gw, acc, s);
    }
    yolo_final<<<1, 1, 0, stream>>>(acc, (float*)d_out);
}